// Model_75084618269276
// MI455X (gfx1250) — compile-verified
//
#include <hip/hip_runtime.h>
#include <math.h>

// ---------------------------------------------------------------------------
// Sequential 131072-step RNN scan (4-wide nonlinear state) + func_net head.
// Latency-bound: single wave32, lane-parallel neurons, v_readlane SGPR
// broadcast between layers, s_load-pipelined x stream. WMMA f32 16x16x4 is
// used for the one-shot func_net head (the only real matrix shapes).
// ---------------------------------------------------------------------------

typedef float v2f __attribute__((ext_vector_type(2)));
typedef float v8f __attribute__((ext_vector_type(8)));

__device__ __forceinline__ float rl(float v, int l) {
  return __int_as_float(__builtin_amdgcn_readlane(__float_as_int(v), l));
}

__device__ __forceinline__ v8f wmma4(v2f a, v2f b, v8f c) {
  // V_WMMA_F32_16X16X4_F32: D = A(16x4) * B(4x16) + C(16x16)
  return __builtin_amdgcn_wmma_f32_16x16x4_f32(false, a, false, b, (short)0, c,
                                               false, false);
}

// A-matrix (16x4 f32) per ISA layout: lanes 0-15 hold M=lane, K = 4c+{0,1};
// lanes 16-31 hold M=lane-16, K = 4c+{2,3}. Rows >= nrows are zero-padded.
__device__ __forceinline__ v2f loadA(const float* __restrict__ W, int ld,
                                     int nrows, int mbase, int kchunk,
                                     int lane) {
  const int row = mbase + (lane & 15);
  const int k = kchunk * 4 + ((lane >> 4) << 1);
  v2f a;
  a.x = (row < nrows) ? W[row * ld + k] : 0.0f;
  a.y = (row < nrows) ? W[row * ld + k + 1] : 0.0f;
  return a;
}

// B-matrix (4x16 f32), every column = the input vector h[4c..4c+3] so that
// every column of D equals W@h. VGPR0 carries rows K0 (lanes 0-15) / K2
// (lanes 16-31); VGPR1 carries K1/K3 (mirror of the A layout).
__device__ __forceinline__ v2f makeB(float h0, float h1, float h2, float h3,
                                     int lane) {
  const bool lo = lane < 16;
  v2f b;
  b.x = lo ? h0 : h2;
  b.y = lo ? h1 : h3;
  return b;
}

__global__ __launch_bounds__(32) void rnn_scan_kernel(
    const float* __restrict__ x,
    const float* __restrict__ r1w0, const float* __restrict__ r1b0,
    const float* __restrict__ r1w1, const float* __restrict__ r1b1,
    const float* __restrict__ r1w2, const float* __restrict__ r1b2,
    const float* __restrict__ r1w3, const float* __restrict__ r1b3,
    const float* __restrict__ r1w4, const float* __restrict__ r1b4,
    const float* __restrict__ h0w, const float* __restrict__ h0b,
    const float* __restrict__ h1w, const float* __restrict__ h1b,
    const float* __restrict__ h2w, const float* __restrict__ h2b,
    const float* __restrict__ h3w, const float* __restrict__ h3b,
    const float* __restrict__ f0w, const float* __restrict__ f0b,
    const float* __restrict__ f1w, const float* __restrict__ f1b,
    const float* __restrict__ f2w, const float* __restrict__ f2b,
    const float* __restrict__ f3w, const float* __restrict__ f3b,
    const float* __restrict__ f4w, const float* __restrict__ f4b,
    float* __restrict__ out, int T) {
  const int lane = threadIdx.x;  // wave32: one wave, lane = neuron index
  const bool a4 = lane < 4, a8 = lane < 8, a12 = lane < 12;

  // ---- preload all recurrent weights into per-lane registers (once) ----
  float w1[8], w2[8], w3[4], w4[4], w5[4];
  float hw0[12], hw1[12], hw2[4], hw3[4];
#pragma unroll
  for (int j = 0; j < 8; ++j) w1[j] = a8 ? r1w0[lane * 8 + j] : 0.0f;
  float b1v = a8 ? r1b0[lane] : 0.0f;
#pragma unroll
  for (int j = 0; j < 8; ++j) w2[j] = a4 ? r1w1[lane * 8 + j] : 0.0f;
  float b2v = a4 ? r1b1[lane] : 0.0f;
#pragma unroll
  for (int j = 0; j < 4; ++j) w3[j] = a4 ? r1w2[lane * 4 + j] : 0.0f;
  float b3v = a4 ? r1b2[lane] : 0.0f;
#pragma unroll
  for (int j = 0; j < 4; ++j) w4[j] = a4 ? r1w3[lane * 4 + j] : 0.0f;
  float b4v = a4 ? r1b3[lane] : 0.0f;
#pragma unroll
  for (int j = 0; j < 4; ++j) w5[j] = a4 ? r1w4[lane * 4 + j] : 0.0f;
  float b5v = a4 ? r1b4[lane] : 0.0f;
#pragma unroll
  for (int j = 0; j < 12; ++j) hw0[j] = a12 ? h0w[lane * 12 + j] : 0.0f;
  float hb0v = a12 ? h0b[lane] : 0.0f;
#pragma unroll
  for (int j = 0; j < 12; ++j) hw1[j] = a4 ? h1w[lane * 12 + j] : 0.0f;
  float hb1v = a4 ? h1b[lane] : 0.0f;
#pragma unroll
  for (int j = 0; j < 4; ++j) hw2[j] = a4 ? h2w[lane * 4 + j] : 0.0f;
  float hb2v = a4 ? h2b[lane] : 0.0f;
#pragma unroll
  for (int j = 0; j < 4; ++j) hw3[j] = a4 ? h3w[lane * 4 + j] : 0.0f;
  float hb3v = a4 ? h3b[lane] : 0.0f;

  // ---- the serial scan ----
  float s0 = 0.f, s1 = 0.f, s2 = 0.f, s3 = 0.f;
  float y0 = 0.f, y1 = 0.f, y2 = 0.f, y3 = 0.f;
  const float4* __restrict__ x4 = (const float4*)x;
  float4 xc = x4[0];

#pragma unroll 1
  for (int t = 0; t < T; ++t) {
    const int tn = (t + 1 < T) ? (t + 1) : (T - 1);
    float4 xn = x4[tn];  // uniform index -> s_load; hides under step body

    // rnn1 L1: Linear(8,8) + sigmoid. input = [x, s]
    float a = b1v;
    a = fmaf(w1[0], xc.x, a); a = fmaf(w1[1], xc.y, a);
    a = fmaf(w1[2], xc.z, a); a = fmaf(w1[3], xc.w, a);
    a = fmaf(w1[4], s0, a);   a = fmaf(w1[5], s1, a);
    a = fmaf(w1[6], s2, a);   a = fmaf(w1[7], s3, a);
    float g = __builtin_amdgcn_rcpf(1.0f + __expf(-a));
    float gg[8];
#pragma unroll
    for (int j = 0; j < 8; ++j) gg[j] = rl(g, j);

    // rnn1 L2: Linear(8,4) + relu
    float c2 = b2v;
#pragma unroll
    for (int j = 0; j < 8; ++j) c2 = fmaf(w2[j], gg[j], c2);
    c2 = fmaxf(c2, 0.0f);
    float r0 = rl(c2, 0), r1 = rl(c2, 1), r2 = rl(c2, 2), r3 = rl(c2, 3);

    // rnn1 L3..L5: Linear(4,4) + relu
    float c3 = b3v;
    c3 = fmaf(w3[0], r0, c3); c3 = fmaf(w3[1], r1, c3);
    c3 = fmaf(w3[2], r2, c3); c3 = fmaf(w3[3], r3, c3);
    c3 = fmaxf(c3, 0.0f);
    float u0 = rl(c3, 0), u1 = rl(c3, 1), u2 = rl(c3, 2), u3 = rl(c3, 3);

    float c4 = b4v;
    c4 = fmaf(w4[0], u0, c4); c4 = fmaf(w4[1], u1, c4);
    c4 = fmaf(w4[2], u2, c4); c4 = fmaf(w4[3], u3, c4);
    c4 = fmaxf(c4, 0.0f);
    float p0 = rl(c4, 0), p1 = rl(c4, 1), p2 = rl(c4, 2), p3 = rl(c4, 3);

    float c5 = b5v;
    c5 = fmaf(w5[0], p0, c5); c5 = fmaf(w5[1], p1, c5);
    c5 = fmaf(w5[2], p2, c5); c5 = fmaf(w5[3], p3, c5);
    c5 = fmaxf(c5, 0.0f);
    y0 = rl(c5, 0) + xc.x; y1 = rl(c5, 1) + xc.y;
    y2 = rl(c5, 2) + xc.z; y3 = rl(c5, 3) + xc.w;

    // hidden L1: Linear(12,12) + relu. input = [S, x, y]
    float ha = hb0v;
    ha = fmaf(hw0[0], s0, ha);   ha = fmaf(hw0[1], s1, ha);
    ha = fmaf(hw0[2], s2, ha);   ha = fmaf(hw0[3], s3, ha);
    ha = fmaf(hw0[4], xc.x, ha); ha = fmaf(hw0[5], xc.y, ha);
    ha = fmaf(hw0[6], xc.z, ha); ha = fmaf(hw0[7], xc.w, ha);
    ha = fmaf(hw0[8], y0, ha);   ha = fmaf(hw0[9], y1, ha);
    ha = fmaf(hw0[10], y2, ha);  ha = fmaf(hw0[11], y3, ha);
    ha = fmaxf(ha, 0.0f);
    float qq[12];
#pragma unroll
    for (int j = 0; j < 12; ++j) qq[j] = rl(ha, j);

    // hidden L2: Linear(12,4) + relu
    float hc = hb1v;
#pragma unroll
    for (int j = 0; j < 12; ++j) hc = fmaf(hw1[j], qq[j], hc);
    hc = fmaxf(hc, 0.0f);
    float d0 = rl(hc, 0), d1 = rl(hc, 1), d2 = rl(hc, 2), d3 = rl(hc, 3);

    // hidden L3: Linear(4,4) + relu
    float he = hb2v;
    he = fmaf(hw2[0], d0, he); he = fmaf(hw2[1], d1, he);
    he = fmaf(hw2[2], d2, he); he = fmaf(hw2[3], d3, he);
    he = fmaxf(he, 0.0f);
    float e0 = rl(he, 0), e1 = rl(he, 1), e2 = rl(he, 2), e3 = rl(he, 3);

    // hidden L4: Linear(4,4) (no activation)
    float hf = hb3v;
    hf = fmaf(hw3[0], e0, hf); hf = fmaf(hw3[1], e1, hf);
    hf = fmaf(hw3[2], e2, hf); hf = fmaf(hw3[3], e3, hf);
    float m0 = rl(hf, 0), m1 = rl(hf, 1), m2 = rl(hf, 2), m3 = rl(hf, 3);

    // carry: s' = s2 + S + y
    s0 = m0 + s0 + y0; s1 = m1 + s1 + y1;
    s2 = m2 + s2 + y2; s3 = m3 + s3 + y3;
    xc = xn;
  }

  // ---- func_net head on final y (runs once) — WMMA f32 16x16x4 chains ----
  const v8f Z = {0.f, 0.f, 0.f, 0.f, 0.f, 0.f, 0.f, 0.f};

  // f0: Linear(4,4) + relu
  v8f D = wmma4(loadA(f0w, 4, 4, 0, 0, lane), makeB(y0, y1, y2, y3, lane), Z);
  float h1_[4];
#pragma unroll
  for (int m = 0; m < 4; ++m) h1_[m] = fmaxf(rl(D[m], 0) + f0b[m], 0.0f);

  // f1: Linear(4,4) + relu
  D = wmma4(loadA(f1w, 4, 4, 0, 0, lane),
            makeB(h1_[0], h1_[1], h1_[2], h1_[3], lane), Z);
  float h2_[4];
#pragma unroll
  for (int m = 0; m < 4; ++m) h2_[m] = fmaxf(rl(D[m], 0) + f1b[m], 0.0f);

  // f2: Linear(4,8) + relu
  D = wmma4(loadA(f2w, 4, 8, 0, 0, lane),
            makeB(h2_[0], h2_[1], h2_[2], h2_[3], lane), Z);
  float h3_[8];
#pragma unroll
  for (int m = 0; m < 8; ++m) h3_[m] = fmaxf(rl(D[m], 0) + f2b[m], 0.0f);

  // f3: Linear(8,32) — two 16-row tiles x two K-chunks, then softmax
  v8f Dt0 = Z, Dt1 = Z;
#pragma unroll
  for (int c = 0; c < 2; ++c) {
    v2f B = makeB(h3_[4 * c], h3_[4 * c + 1], h3_[4 * c + 2], h3_[4 * c + 3],
                  lane);
    Dt0 = wmma4(loadA(f3w, 8, 32, 0, c, lane), B, Dt0);
    Dt1 = wmma4(loadA(f3w, 8, 32, 16, c, lane), B, Dt1);
  }
  float z[32];
#pragma unroll
  for (int m = 0; m < 16; ++m)
    z[m] = rl(Dt0[m & 7], (m < 8) ? 0 : 16) + f3b[m];
#pragma unroll
  for (int m = 0; m < 16; ++m)
    z[16 + m] = rl(Dt1[m & 7], (m < 8) ? 0 : 16) + f3b[16 + m];

  float mx = z[0];
#pragma unroll
  for (int m = 1; m < 32; ++m) mx = fmaxf(mx, z[m]);
  float prob[32], sum = 0.0f;
#pragma unroll
  for (int m = 0; m < 32; ++m) { prob[m] = __expf(z[m] - mx); sum += prob[m]; }
#pragma unroll
  for (int m = 0; m < 32; ++m) prob[m] /= sum;

  // f4: Linear(32,4) — 8 chained K-chunks
  v8f D4 = Z;
#pragma unroll
  for (int c = 0; c < 8; ++c) {
    v2f B = makeB(prob[4 * c], prob[4 * c + 1], prob[4 * c + 2],
                  prob[4 * c + 3], lane);
    D4 = wmma4(loadA(f4w, 32, 4, 0, c, lane), B, D4);
  }
  if (lane == 0) {
#pragma unroll
    for (int m = 0; m < 4; ++m) out[m] = D4[m] + f4b[m];  // lane0 = column N=0
  }
}

extern "C" void kernel_launch(void* const* d_in, const int* in_sizes, int n_in,
                              void* d_out, int out_size, void* d_ws,
                              size_t ws_size, hipStream_t stream) {
  const float* p[29];
  for (int i = 0; i < 29 && i < n_in; ++i) p[i] = (const float*)d_in[i];
  const int T = in_sizes[0] / 4;
  rnn_scan_kernel<<<1, 32, 0, stream>>>(
      p[0],
      p[1], p[2], p[3], p[4], p[5], p[6], p[7], p[8], p[9], p[10],
      p[11], p[12], p[13], p[14], p[15], p[16], p[17], p[18],
      p[19], p[20], p[21], p[22], p[23], p[24], p[25], p[26], p[27], p[28],
      (float*)d_out, T);
}